// InnerProductPredictor_16295105921290
// MI455X (gfx1250) — compile-verified
//
#include <hip/hip_runtime.h>
#include <stdint.h>

typedef __attribute__((ext_vector_type(2))) float v2f;
typedef __attribute__((ext_vector_type(8))) float v8f;

#define NUM_CELLS 64
#define D 128
#define TILE 16
#define PITCH 132   // LDS row pitch in floats (padded: bank = (4*row + col) % 64)

// ---------------- sort-by-cell-line pipeline ----------------

__global__ void zero_kernel(int* p, int n) {
    int i = blockIdx.x * blockDim.x + threadIdx.x;
    if (i < n) p[i] = 0;
}

__global__ void histo_kernel(const long long* cells, int* counts, int n) {
    int i = blockIdx.x * blockDim.x + threadIdx.x;
    if (i < n) atomicAdd(&counts[(int)cells[i]], 1);
}

__global__ void scan_kernel(const int* counts, int* offsets) {
    if (threadIdx.x == 0) {
        int s = 0;
        for (int c = 0; c < NUM_CELLS; ++c) { offsets[c] = s; s += counts[c]; }
    }
}

__global__ void scatter_kernel(const long long* cells, const int* offsets,
                               int* cursors, int* order, int n) {
    int i = blockIdx.x * blockDim.x + threadIdx.x;
    if (i < n) {
        int c = (int)cells[i];
        int pos = offsets[c] + atomicAdd(&cursors[c], 1);
        order[pos] = i;
    }
}

// ---------------- main WMMA kernel ----------------
// Block = (cell c, tile t): 16 samples of cell c. 256 threads = 8 waves.
// Wave w computes output columns [16w, 16w+16) of H1, H2, T.

__global__ __launch_bounds__(256)
void predict_kernel(const long long* __restrict__ drug_pairs,
                    const float* __restrict__ h_drug,
                    const float* __restrict__ pred_lin,
                    const float* __restrict__ pred_prod,
                    const int* __restrict__ counts,
                    const int* __restrict__ offsets,
                    const int* __restrict__ order,
                    float* __restrict__ out)
{
    const int c   = blockIdx.x;
    const int t   = blockIdx.y;
    const int cnt = counts[c];
    if (t * TILE >= cnt) return;

    __shared__ float X1s[TILE * PITCH];
    __shared__ float X2s[TILE * PITCH];
    __shared__ float H1s[TILE * PITCH];
    __shared__ float H2s[TILE * PITCH];
    __shared__ int   rowIdx[TILE];
    __shared__ float partial[TILE];

    const int tid  = threadIdx.x;
    const int lane = tid & 31;
    const int wave = tid >> 5;        // 0..7
    const int half = lane >> 4;       // 0/1
    const int l16  = lane & 15;
    const int ncol = wave * 16 + l16; // this lane's output column

    if (tid < TILE) {
        int g = t * TILE + tid;
        rowIdx[tid]  = (g < cnt) ? order[offsets[c] + g] : -1;
        partial[tid] = 0.0f;
    }
    __syncthreads();

    // ---- stage X1, X2 tiles (16 x 128 each); zero-pad invalid rows ----
    {
        int r  = tid >> 4;          // row 0..15
        int c8 = (tid & 15) * 8;    // 8 floats per thread per matrix
        int b  = rowIdx[r];
        if (b >= 0) {
            long long d1 = drug_pairs[2 * (long long)b + 0];
            long long d2 = drug_pairs[2 * (long long)b + 1];
            const float* p1 = h_drug + (size_t)d1 * D + c8;
            const float* p2 = h_drug + (size_t)d2 * D + c8;
            #pragma unroll
            for (int j = 0; j < 8; ++j) {
                X1s[r * PITCH + c8 + j] = p1[j];
                X2s[r * PITCH + c8 + j] = p2[j];
            }
        } else {
            #pragma unroll
            for (int j = 0; j < 8; ++j) {
                X1s[r * PITCH + c8 + j] = 0.0f;
                X2s[r * PITCH + c8 + j] = 0.0f;
            }
        }
    }
    __syncthreads();

    const float* Wc = pred_lin  + (size_t)c * D * D;
    const float* Pc = pred_prod + (size_t)c * D * D;

    // ---- H1 = relu(X1 @ W), H2 = relu(X2 @ W); share the W fragment ----
    v8f acc1 = {};
    v8f acc2 = {};
    #pragma unroll 4
    for (int kk = 0; kk < D / 4; ++kk) {
        int k0 = kk * 4 + 2 * half;      // this lane's K pair within the K=4 step
        v2f a1, a2, bw;
        a1.x = X1s[l16 * PITCH + k0];
        a1.y = X1s[l16 * PITCH + k0 + 1];
        a2.x = X2s[l16 * PITCH + k0];
        a2.y = X2s[l16 * PITCH + k0 + 1];
        bw.x = Wc[(size_t)k0 * D + ncol];
        bw.y = Wc[(size_t)(k0 + 1) * D + ncol];
        acc1 = __builtin_amdgcn_wmma_f32_16x16x4_f32(false, a1, false, bw,
                                                     (short)0, acc1, false, false);
        acc2 = __builtin_amdgcn_wmma_f32_16x16x4_f32(false, a2, false, bw,
                                                     (short)0, acc2, false, false);
    }
    #pragma unroll
    for (int v = 0; v < 8; ++v) {
        int m = v + 8 * half;            // C/D layout: VGPR v, half-wave -> row
        H1s[m * PITCH + ncol] = fmaxf(acc1[v], 0.0f);
        H2s[m * PITCH + ncol] = fmaxf(acc2[v], 0.0f);
    }
    __syncthreads();

    // ---- T = H1 @ (P + P^T) ----
    v8f accT = {};
    #pragma unroll 4
    for (int kk = 0; kk < D / 4; ++kk) {
        int k0 = kk * 4 + 2 * half;
        v2f a, bg;
        a.x = H1s[l16 * PITCH + k0];
        a.y = H1s[l16 * PITCH + k0 + 1];
        bg.x = Pc[(size_t)k0 * D + ncol]       + Pc[(size_t)ncol * D + k0];
        bg.y = Pc[(size_t)(k0 + 1) * D + ncol] + Pc[(size_t)ncol * D + (k0 + 1)];
        accT = __builtin_amdgcn_wmma_f32_16x16x4_f32(false, a, false, bg,
                                                     (short)0, accT, false, false);
    }

    // ---- out[m] = sum_n T[m][n] * H2[m][n]; reduce across 16 lanes, then waves ----
    #pragma unroll
    for (int v = 0; v < 8; ++v) {
        int m = v + 8 * half;
        float r = accT[v] * H2s[m * PITCH + ncol];
        r += __shfl_xor(r, 1, 32);
        r += __shfl_xor(r, 2, 32);
        r += __shfl_xor(r, 4, 32);
        r += __shfl_xor(r, 8, 32);       // sum within each 16-lane half
        if (l16 == 0) atomicAdd(&partial[m], r);   // ds_add_f32 across waves
    }
    __syncthreads();

    if (tid < TILE) {
        int b = rowIdx[tid];
        if (b >= 0) out[b] = partial[tid];
    }
}

// ---------------- host launch ----------------

extern "C" void kernel_launch(void* const* d_in, const int* in_sizes, int n_in,
                              void* d_out, int out_size, void* d_ws, size_t ws_size,
                              hipStream_t stream) {
    const long long* drug_pairs = (const long long*)d_in[0];
    const long long* cell_lines = (const long long*)d_in[1];
    const float*     h_drug     = (const float*)d_in[2];
    const float*     pred_lin   = (const float*)d_in[3];
    const float*     pred_prod  = (const float*)d_in[4];
    float*           out        = (float*)d_out;

    const int B = in_sizes[1];           // number of batch elements (cell_lines)

    // workspace: counts[64] | offsets[64] | cursors[64] | order[B]
    int* counts  = (int*)d_ws;
    int* offsets = counts + 64;
    int* cursors = counts + 128;
    int* order   = counts + 192;

    zero_kernel<<<1, 256, 0, stream>>>(counts, 192);
    histo_kernel<<<(B + 255) / 256, 256, 0, stream>>>(cell_lines, counts, B);
    scan_kernel<<<1, 32, 0, stream>>>(counts, offsets);
    scatter_kernel<<<(B + 255) / 256, 256, 0, stream>>>(cell_lines, offsets, cursors, order, B);

    const int max_tiles = (B + TILE - 1) / TILE;
    dim3 grid(NUM_CELLS, max_tiles);
    predict_kernel<<<grid, 256, 0, stream>>>(drug_pairs, h_drug, pred_lin, pred_prod,
                                             counts, offsets, order, out);
}